// BandSplitEncoder_27135603376305
// MI455X (gfx1250) — compile-verified
//
#include <hip/hip_runtime.h>

typedef __attribute__((ext_vector_type(2))) float v2f;
typedef __attribute__((ext_vector_type(8))) float v8f;

#define N_BANDS 62
#define TOTAL_C 2048
#define N_DIM   512
#define N_ROWS  2048   /* 4 * 512 */
#define EPS     1e-12f

// Band table: 24x4, 12x8, 8x24, 8x48, 8x96, 2x256  (offsets 0,96,192,384,768,1536)
__device__ __forceinline__ void band_info(int band, int* d, int* off) {
  if (band < 24)      { *d = 4;   *off = band * 4; }
  else if (band < 36) { *d = 8;   *off = 96   + (band - 24) * 8; }
  else if (band < 44) { *d = 24;  *off = 192  + (band - 36) * 24; }
  else if (band < 52) { *d = 48;  *off = 384  + (band - 44) * 48; }
  else if (band < 60) { *d = 96;  *off = 768  + (band - 52) * 96; }
  else                { *d = 256; *off = 1536 + (band - 60) * 256; }
}

// Kernel 1: per (row, band) normalization scale = sqrt(d) / max(||x_band||, eps)
__global__ void band_scale_kernel(const float* __restrict__ x,
                                  float* __restrict__ scales) {
  int idx = blockIdx.x * blockDim.x + threadIdx.x;
  if (idx >= N_ROWS * N_BANDS) return;
  int row  = idx / N_BANDS;
  int band = idx - row * N_BANDS;
  int d, off;
  band_info(band, &d, &off);
  const float* xr = x + (size_t)row * TOTAL_C + off;
  float ss = 0.f;
  for (int k = 0; k < d; ++k) { float v = xr[k]; ss = fmaf(v, v, ss); }
  float nrm = sqrtf(ss);
  scales[idx] = sqrtf((float)d) / fmaxf(nrm, EPS);
}

// Kernel 2: per (band, 16-row M tile) block; 8 waves x (4 WMMA N-tiles of 16)
// covering the full N=512. fp32 WMMA 16x16x4, K stepped by 4.
__global__ void __launch_bounds__(256)
band_gemm_kernel(const float* __restrict__ x, const float* __restrict__ gamma,
                 const float* __restrict__ W, const float* __restrict__ bias,
                 const float* __restrict__ scales, float* __restrict__ out) {
  const int mtile = blockIdx.x;   // 0..127  (M tile of 16 rows)
  const int band  = blockIdx.y;   // 0..61
  int d, off;
  band_info(band, &d, &off);

  const int lane  = threadIdx.x & 31;
  const int wave  = threadIdx.x >> 5;       // 0..7
  const int nbase = wave * 64;              // wave's N window
  const int lan15 = lane & 15;
  const int khalf = (lane >> 4) * 2;        // A/B: lanes 16-31 hold K+2..K+3

  const int mrow = mtile * 16 + lan15;      // A-fragment row (ISA 32b A 16x4 layout)
  const float sc = scales[mrow * N_BANDS + band];

  // C init = bias: every C VGPR of a lane maps to the same N column.
  v8f acc[4];
#pragma unroll
  for (int t = 0; t < 4; ++t) {
    float bv = bias[band * N_DIM + nbase + t * 16 + lan15];
    acc[t] = (v8f){bv, bv, bv, bv, bv, bv, bv, bv};
  }

  const float* xrow = x + (size_t)mrow * TOTAL_C + off;

  for (int kk = 0; kk < d; kk += 4) {
    const int kg = kk + khalf;                       // even -> 8B aligned
    const float2 av = *(const float2*)(xrow + kg);
    const float2 gv = *(const float2*)(gamma + off + kg);
    v2f a;
    a.x = av.x * gv.x * sc;
    a.y = av.y * gv.y * sc;

    const float* wr0 = W + (size_t)(off + kg) * N_DIM;  // K row kg
    const float* wr1 = wr0 + N_DIM;                     // K row kg+1
#pragma unroll
    for (int t = 0; t < 4; ++t) {
      const int n = nbase + t * 16 + lan15;
      v2f bf;
      bf.x = wr0[n];
      bf.y = wr1[n];
      acc[t] = __builtin_amdgcn_wmma_f32_16x16x4_f32(
          /*neg_a=*/false, a, /*neg_b=*/false, bf,
          /*c_mod=*/(short)0, acc[t], /*reuse_a=*/false, /*reuse_b=*/false);
    }
  }

  // C/D layout: VGPR v -> M = v + (lane>=16 ? 8 : 0); N = lane&15 (+ tile base).
  // Output is write-once and > L2 (260 MB): use non-temporal stores.
  const int mv = mtile * 16 + (lane >> 4) * 8;
#pragma unroll
  for (int t = 0; t < 4; ++t) {
    const int n = nbase + t * 16 + lan15;
#pragma unroll
    for (int v = 0; v < 8; ++v) {
      size_t o = ((size_t)(mv + v) * N_BANDS + band) * N_DIM + n;
      __builtin_nontemporal_store(acc[t][v], &out[o]);
    }
  }
}

extern "C" void kernel_launch(void* const* d_in, const int* in_sizes, int n_in,
                              void* d_out, int out_size, void* d_ws, size_t ws_size,
                              hipStream_t stream) {
  const float* x     = (const float*)d_in[0];  // (4, 512, 2048)
  const float* gamma = (const float*)d_in[1];  // (2048,)
  const float* W     = (const float*)d_in[2];  // (2048, 512)
  const float* b     = (const float*)d_in[3];  // (62, 512)
  float* out    = (float*)d_out;               // (4, 512, 62, 512)
  float* scales = (float*)d_ws;                // 2048*62 floats (~0.5 MB)

  const int total = N_ROWS * N_BANDS;
  band_scale_kernel<<<(total + 255) / 256, 256, 0, stream>>>(x, scales);

  dim3 grid(128, 62);
  band_gemm_kernel<<<grid, 256, 0, stream>>>(x, gamma, W, b, scales, out);
}